// Dynamic_Anchor_Layer_30923764531484
// MI455X (gfx1250) — compile-verified
//
#include <hip/hip_runtime.h>
#include <math.h>

#define BATCH 2048
#define CH    32
#define LEN   500
#define NBINS 251      // rfft bins 0..250
#define NF    249      // analysis bins 1..249
#define NCOLP 256      // padded frequency columns (WMMA-friendly)
#define NPAIR 250      // K-pairs (LEN/2) for float2-interleaved twiddles

typedef float v2f __attribute__((ext_vector_type(2)));
typedef float v8f __attribute__((ext_vector_type(8)));

// ---------------------------------------------------------------------------
// Kernel 1: DFT twiddle tables, interleaved as float2 along the time axis n so
// WMMA A/B fragments are single b64 loads.
//   wc2[np*NCOLP + col] = { cos(2*pi*(2np)*col/LEN),  cos(2*pi*(2np+1)*col/LEN) }
//   ws2[np*NCOLP + col] = {-sin(2*pi*(2np)*col/LEN), -sin(2*pi*(2np+1)*col/LEN) }
// Columns 251..255 are zero (padding; their outputs are never stored).
// ---------------------------------------------------------------------------
__global__ void __launch_bounds__(256) build_twiddles(float* __restrict__ wc2,
                                                      float* __restrict__ ws2) {
    const int gid = blockIdx.x * 256 + threadIdx.x;   // [0, NPAIR*NCOLP)
    const int np  = gid >> 8;
    const int col = gid & 255;
    float c0 = 0.f, s0 = 0.f, c1 = 0.f, s1 = 0.f;
    if (col < NBINS) {
        const int n0 = 2 * np, n1 = 2 * np + 1;
        const int m0 = (n0 * col) % LEN;
        const int m1 = (n1 * col) % LEN;
        const float k = -6.28318530717958647692f / (float)LEN;
        __sincosf((float)m0 * k, &s0, &c0);
        __sincosf((float)m1 * k, &s1, &c1);
    }
    const size_t o = 2 * (size_t)gid;
    wc2[o] = c0; wc2[o + 1] = c1;
    ws2[o] = s0; ws2[o + 1] = s1;
}

// ---------------------------------------------------------------------------
// Kernel 2: PSD via fp32 WMMA GEMM.  Block = 256 threads = 8 waves.
// Block tile: 32 rows x 256 cols.  Wave: 16 rows x 64 cols -> 4 col-tiles x
// {cos,sin} = 8 v8f accumulators, updated strictly in place (D == C) inside a
// non-unrolled K loop so the register allocator keeps one fixed accumulator
// set (avoids the v_dual_mov accumulator-rotation seen with unroll-by-2).
//   A fragment (16x4 f32): lanes 0-15 hold K={k0,k0+1}, lanes 16-31 K={k0+2,k0+3}.
//   B fragment (4x16 f32): v0 lanes0-15 K=k0 / lanes16-31 K=k0+2; v1: k0+1/k0+3.
// ---------------------------------------------------------------------------
__global__ void __launch_bounds__(256) dft_psd(const float* __restrict__ x,
                                               const float* __restrict__ wc2,
                                               const float* __restrict__ ws2,
                                               float* __restrict__ fd) {
    const int tid   = threadIdx.x;
    const int wave  = tid >> 5;
    const int lane  = tid & 31;
    const int half  = lane >> 4;   // 0: lanes 0-15, 1: lanes 16-31
    const int l16   = lane & 15;
    const int mtile = wave >> 2;   // 0..1
    const int ngrp  = wave & 3;    // 0..3

    const int rowBase = blockIdx.x * 32 + mtile * 16;
    const int colBase = ngrp * 64;

    const v8f zero = {};
    v8f c0a = zero, c1a = zero, c2a = zero, c3a = zero;   // cos accumulators
    v8f s0a = zero, s1a = zero, s2a = zero, s3a = zero;   // sin accumulators

    // A pointer: row (rowBase+l16), element (k0 + 2*half); bumps by 4 floats/step
    const float* px = x + (size_t)(rowBase + l16) * LEN + 2 * half;
    // B pointers: float2-interleaved rows; np = k0/2 + half advances by 2/step
    const float* pc = wc2 + 2 * ((size_t)half * NCOLP + colBase + l16);
    const float* ps = ws2 + 2 * ((size_t)half * NCOLP + colBase + l16);

#pragma unroll 1
    for (int step = 0; step < LEN / 4; ++step) {
        const v2f a   = *(const v2f*)px;
        const v2f bc0 = *(const v2f*)(pc);
        const v2f bc1 = *(const v2f*)(pc + 32);
        const v2f bc2 = *(const v2f*)(pc + 64);
        const v2f bc3 = *(const v2f*)(pc + 96);
        const v2f bs0 = *(const v2f*)(ps);
        const v2f bs1 = *(const v2f*)(ps + 32);
        const v2f bs2 = *(const v2f*)(ps + 64);
        const v2f bs3 = *(const v2f*)(ps + 96);

        c0a = __builtin_amdgcn_wmma_f32_16x16x4_f32(false, a, false, bc0, (short)0, c0a, false, false);
        s0a = __builtin_amdgcn_wmma_f32_16x16x4_f32(false, a, false, bs0, (short)0, s0a, false, false);
        c1a = __builtin_amdgcn_wmma_f32_16x16x4_f32(false, a, false, bc1, (short)0, c1a, false, false);
        s1a = __builtin_amdgcn_wmma_f32_16x16x4_f32(false, a, false, bs1, (short)0, s1a, false, false);
        c2a = __builtin_amdgcn_wmma_f32_16x16x4_f32(false, a, false, bc2, (short)0, c2a, false, false);
        s2a = __builtin_amdgcn_wmma_f32_16x16x4_f32(false, a, false, bs2, (short)0, s2a, false, false);
        c3a = __builtin_amdgcn_wmma_f32_16x16x4_f32(false, a, false, bc3, (short)0, c3a, false, false);
        s3a = __builtin_amdgcn_wmma_f32_16x16x4_f32(false, a, false, bs3, (short)0, s3a, false, false);

        px += 4;                    // next 4 time samples
        pc += 2 * 2 * NCOLP;        // np += 2 -> 1024 floats
        ps += 2 * 2 * NCOLP;
    }

    // Epilogue: PSD = Re^2 + Im^2; C/D layout: vgpr i -> row rowBase + i + 8*half
    const int colL = colBase + l16;
    const size_t rb = (size_t)(rowBase + 8 * half) * NBINS;
#pragma unroll
    for (int i = 0; i < 8; ++i) {
        const size_t ro = rb + (size_t)i * NBINS;
        if (colL < NBINS)      fd[ro + colL]      = c0a[i] * c0a[i] + s0a[i] * s0a[i];
        if (colL + 16 < NBINS) fd[ro + colL + 16] = c1a[i] * c1a[i] + s1a[i] * s1a[i];
        if (colL + 32 < NBINS) fd[ro + colL + 32] = c2a[i] * c2a[i] + s2a[i] * s2a[i];
        if (colL + 48 < NBINS) fd[ro + colL + 48] = c3a[i] * c3a[i] + s3a[i] * s3a[i];
    }
}

// ---------------------------------------------------------------------------
// Kernel 3: peak analysis + band construction, one block per batch element.
// ---------------------------------------------------------------------------
__global__ void __launch_bounds__(256) analyze_bands(const float* __restrict__ fd,
                                                     float* __restrict__ bands) {
    const int b   = blockIdx.x;
    const int tid = threadIdx.x;

    __shared__ float fm[NF];
    __shared__ float sm[NF];
    __shared__ int   vld[NF];

    // channel mean of PSD bins 1..249
    if (tid < NF) {
        const float* base = fd + (size_t)b * CH * NBINS + (tid + 1);
        float s = 0.f;
#pragma unroll 4
        for (int c = 0; c < CH; ++c) s += base[(size_t)c * NBINS];
        fm[tid] = s * (1.0f / CH);
    }
    __syncthreads();

    // gaussian smoothing: kernel [w0,w1,w0], w ~ exp(-k^2/(2*sigma^2)), sigma=2
    if (tid < NF) {
        const float e  = __expf(-0.125f);
        const float nm = 1.0f / (1.0f + 2.0f * e);
        const float w0 = e * nm, w1 = nm;
        const float left  = (tid > 0)      ? fm[tid - 1] : 0.f;
        const float right = (tid < NF - 1) ? fm[tid + 1] : 0.f;
        sm[tid] = w0 * left + w1 * fm[tid] + w0 * right;
    }
    __syncthreads();

    // peak flag + min-distance suppression (3 Hz / 0.256 Hz -> within-window = +-11 bins)
    if (tid < NF) {
        int ok = 0;
        if (tid >= 1 && tid <= NF - 2)
            ok = (sm[tid] - sm[tid - 1] > 0.f) && (sm[tid + 1] - sm[tid] < 0.f);
        if (ok) {
            int lo = tid - 11; if (lo < 0) lo = 0;
            int hi = tid + 11; if (hi > NF - 1) hi = NF - 1;
            for (int m = lo; m <= hi; ++m) {
                if (m == tid) continue;
                // ranked stronger: larger value, or equal value at smaller index
                if (sm[m] > sm[tid] || (sm[m] == sm[tid] && m < tid)) { ok = 0; break; }
            }
        }
        vld[tid] = ok;
    }
    __syncthreads();

    if (tid == 0) {
        int best[3] = { -1, -1, -1 };
        int found = 0;
        for (int pass = 0; pass < 3; ++pass) {
            float bv = 0.f; int bi = -1;
            for (int j = 0; j < NF; ++j) {
                if (!vld[j]) continue;
                if (j == best[0] || j == best[1] || j == best[2]) continue;
                if (bi < 0 || sm[j] > bv) { bv = sm[j]; bi = j; }  // strict > : stable
            }
            if (bi < 0) break;
            best[found++] = bi;
        }
        // deterministic fallback for missing peaks (reference uses a fixed PRNG)
        for (int s = found; s < 3; ++s) {
            unsigned r = 8989u + (unsigned)b * 2654435761u + (unsigned)s * 40503u;
            r ^= r >> 16; r *= 2246822519u; r ^= r >> 13;
            best[s] = (int)(r % (unsigned)NF);
        }
        const float df = 128.0f / 500.0f;
        float f0 = (best[0] + 1) * df;
        float f1 = (best[1] + 1) * df;
        float f2 = (best[2] + 1) * df;
        float t;
        if (f0 > f1) { t = f0; f0 = f1; f1 = t; }
        if (f1 > f2) { t = f1; f1 = f2; f2 = t; }
        if (f0 > f1) { t = f0; f0 = f1; f1 = t; }
        const float c1 = 0.5f * (f0 + f1);
        const float c2 = 0.5f * (f1 + f2);
        float* o = bands + (size_t)b * 6;
        o[0] = 0.5f; o[1] = c1; o[2] = c1; o[3] = c2; o[4] = c2; o[5] = 45.0f;
    }
}

// ---------------------------------------------------------------------------
extern "C" void kernel_launch(void* const* d_in, const int* in_sizes, int n_in,
                              void* d_out, int out_size, void* d_ws, size_t ws_size,
                              hipStream_t stream) {
    const float* x   = (const float*)d_in[0];        // [2048, 32, 500] f32
    float* out       = (float*)d_out;
    float* bands     = out;                          // [2048, 3, 2]
    float* fd        = out + (size_t)BATCH * 6;      // [2048, 32, 251]
    float* wc2       = (float*)d_ws;                 // 250*256 float2
    float* ws2       = wc2 + (size_t)NPAIR * NCOLP * 2;

    build_twiddles<<<NPAIR, 256, 0, stream>>>(wc2, ws2);
    dft_psd<<<BATCH, 256, 0, stream>>>(x, wc2, ws2, fd);   // 2048 blocks * 32 rows
    analyze_bands<<<BATCH, 256, 0, stream>>>(fd, bands);
}